// TensorComposerMobius_42829413875989
// MI455X (gfx1250) — compile-verified
//
#include <hip/hip_runtime.h>
#include <hip/hip_bf16.h>
#include <math.h>

// MI455X / gfx1250, wave32. f32 WMMA (16x16x4) for the two true GEMM stages,
// bandwidth-optimal single pass over `projections` in the final stage with
// 8-row h-tiling + LDS staging of gJ to kill redundant L2 streaming.

typedef __attribute__((ext_vector_type(2))) float v2f;
typedef __attribute__((ext_vector_type(8))) float v8f;

#define WMMA_F32_4(A, B, C) \
  __builtin_amdgcn_wmma_f32_16x16x4_f32(false, (A), false, (B), (short)0, (C), false, false)

// ---------------------------------------------------------------------------
// k1: xt[b][n] = sum_k x[b][k] * Wc[n][k] + bc[n]   (B=4 padded to M=16)
// One wave per 16-column tile (64 waves). K=2048 in steps of 4.
// ---------------------------------------------------------------------------
__global__ void k1_gemm_wmma(const float* __restrict__ x,   // [4][2048]
                             const float* __restrict__ Wc,  // [1024][2048]
                             const float* __restrict__ bc,  // [1024]
                             float* __restrict__ xt)        // [4][1024]
{
  const int lane  = threadIdx.x & 31;
  const int wid   = (blockIdx.x * blockDim.x + threadIdx.x) >> 5;  // 0..63
  const int n0    = wid * 16;
  const int m     = lane & 15;     // A row (batch), valid if m<4
  const int khalf = lane >> 4;     // lanes 16-31 carry K+2
  const int n     = lane & 15;     // B/D column
  const bool mv   = (m < 4);
  const int mrow  = (m & 3) * 2048;

  v8f acc = {};
  for (int k0 = 0; k0 < 2048; k0 += 4) {
    const int ks = k0 + 2 * khalf;
    v2f a, b;
    float a0 = x[mrow + ks], a1 = x[mrow + ks + 1];
    a.x = mv ? a0 : 0.0f;
    a.y = mv ? a1 : 0.0f;
    b.x = Wc[(n0 + n) * 2048 + ks];
    b.y = Wc[(n0 + n) * 2048 + ks + 1];
    acc = WMMA_F32_4(a, b, acc);
  }
  if (khalf == 0) {               // rows M=0..7 live in lanes 0-15; keep M<4
    const float bias = bc[n0 + n];
#pragma unroll
    for (int v = 0; v < 4; ++v)
      xt[v * 1024 + n0 + n] = acc[v] + bias;
  }
}

// ---------------------------------------------------------------------------
// k2: xm = xt[:, :512] * siglog(xt[:, 512:])  ->  xr, xi  [4][256]
// ---------------------------------------------------------------------------
__device__ __forceinline__ float siglog(float v) {
  return copysignf(log1pf(fabsf(v)), v);
}

__global__ void k2_siglog(const float* __restrict__ xt,
                          float* __restrict__ xr, float* __restrict__ xi)
{
  const int tid = blockIdx.x * blockDim.x + threadIdx.x;  // 0..1023
  const int b = tid >> 8, l = tid & 255;
  const float a0 = xt[b * 1024 + 2 * l];
  const float a1 = xt[b * 1024 + 2 * l + 1];
  const float g0 = xt[b * 1024 + 512 + 2 * l];
  const float g1 = xt[b * 1024 + 512 + 2 * l + 1];
  xr[b * 256 + l] = a0 * siglog(g0);
  xi[b * 256 + l] = a1 * siglog(g1);
}

// ---------------------------------------------------------------------------
// k3: complex einsum 'bl,tclr->btcr' via WMMA, fused with the collapsed
// broadcast/mag epilogue:  trans = (Re,Im) * 256*mag / (256*mag + EPS).
// One wave per (t,c) pair: 544 waves, K=256, 4 WMMAs per K-step.
// ---------------------------------------------------------------------------
__global__ void k3_einsum_wmma(const float* __restrict__ xr,  // [4][256]
                               const float* __restrict__ xi,  // [4][256]
                               const float* __restrict__ GR,  // [34][16][256][16]
                               const float* __restrict__ GI,  // [34][16][256][16]
                               float* __restrict__ trans)     // [4][34][16][16][2]
{
  const int lane  = threadIdx.x & 31;
  const int wid   = (blockIdx.x * blockDim.x + threadIdx.x) >> 5;  // 0..543
  const int t     = wid >> 4;
  const int c     = wid & 15;
  const int m     = lane & 15;
  const int khalf = lane >> 4;
  const int r     = lane & 15;
  const bool mv   = (m < 4);
  const int mrow  = (m & 3) * 256;

  const float* __restrict__ GRb = GR + (size_t)(t * 16 + c) * 256 * 16;
  const float* __restrict__ GIb = GI + (size_t)(t * 16 + c) * 256 * 16;

  v8f accre = {}, accim = {};
  for (int k0 = 0; k0 < 256; k0 += 4) {
    const int ks = k0 + 2 * khalf;
    float r0 = xr[mrow + ks], r1 = xr[mrow + ks + 1];
    float i0 = xi[mrow + ks], i1 = xi[mrow + ks + 1];
    v2f axr, axi, nxi, bgr, bgi;
    axr.x = mv ? r0 : 0.0f;  axr.y = mv ? r1 : 0.0f;
    axi.x = mv ? i0 : 0.0f;  axi.y = mv ? i1 : 0.0f;
    nxi.x = -axi.x;          nxi.y = -axi.y;
    bgr.x = GRb[ks * 16 + r];       bgr.y = GRb[(ks + 1) * 16 + r];
    bgi.x = GIb[ks * 16 + r];       bgi.y = GIb[(ks + 1) * 16 + r];
    accre = WMMA_F32_4(axr, bgr, accre);   // + xr*GR
    accre = WMMA_F32_4(nxi, bgi, accre);   // - xi*GI
    accim = WMMA_F32_4(axr, bgi, accim);   // + xr*GI
    accim = WMMA_F32_4(axi, bgr, accim);   // + xi*GR
  }
  if (khalf == 0) {
#pragma unroll
    for (int v = 0; v < 4; ++v) {          // batch rows b = v (0..3)
      const float Re = accre[v], Im = accim[v];
      const float mag = sqrtf(Re * Re + Im * Im + 1e-12f);
      const float ms  = 256.0f * mag;
      const float s   = ms / (ms + 1e-12f);
      const int idx = ((((v * 34 + t) * 16 + c) * 16) + r) * 2;
      trans[idx]     = Re * s;
      trans[idx + 1] = Im * s;
    }
  }
}

// ---------------------------------------------------------------------------
// k4: tRe/tIm = complex x @ w_mix ; beta = softmax over R of |t|.
// grid.x = C (16), block = 64 (b = tid/16, r = tid%16), 16-lane shfl softmax.
// ---------------------------------------------------------------------------
__global__ void k4_beta(const float* __restrict__ xr, const float* __restrict__ xi,
                        const float* __restrict__ wm,   // [256][256][2]
                        float* __restrict__ beta)       // [4][16][16]
{
  const int c = blockIdx.x;
  const int tid = threadIdx.x;
  const int b = tid >> 4, r = tid & 15;
  const int k = c * 16 + r;
  float tRe = 0.0f, tIm = 0.0f;
  for (int l = 0; l < 256; ++l) {
    const float w0 = wm[(l * 256 + k) * 2];
    const float w1 = wm[(l * 256 + k) * 2 + 1];
    const float xrl = xr[b * 256 + l];
    const float xil = xi[b * 256 + l];
    tRe += xrl * w0 - xil * w1;
    tIm += xrl * w1 + xil * w0;
  }
  const float s = sqrtf(tRe * tRe + tIm * tIm + 1e-12f);
  float mx = s;
#pragma unroll
  for (int msk = 1; msk < 16; msk <<= 1)
    mx = fmaxf(mx, __shfl_xor(mx, msk, 16));
  const float e = expf(s - mx);
  float sum = e;
#pragma unroll
  for (int msk = 1; msk < 16; msk <<= 1)
    sum += __shfl_xor(sum, msk, 16);
  beta[(b * 16 + c) * 16 + r] = e / sum;
}

// ---------------------------------------------------------------------------
// k5: transform_space + cmul/cdiv with inversions + beta-weighted sum over R.
// mode 0 -> gI (cmul with inv), mode 1 -> gJ (cdiv inv/sj, eps=1e-4).
// grid = (768/256, B*C); thread per (b,c,h).
// ---------------------------------------------------------------------------
__global__ void k5_spaces(const float* __restrict__ space,  // [16][16][768][2]
                          const float* __restrict__ trans,  // [4][34][16][16][2]
                          const float* __restrict__ beta,   // [4][16][16]
                          const float* __restrict__ inv,    // [16][2]
                          float* __restrict__ gout,         // [4][16][768][2]
                          int mode)
{
  const int h  = blockIdx.x * blockDim.x + threadIdx.x;  // 0..767
  const int bc = blockIdx.y;
  const int b = bc >> 4, c = bc & 15;

  // linear interp 16 -> 768 (align_corners=False)
  float pos = (h + 0.5f) * (16.0f / 768.0f) - 0.5f;
  pos = fminf(fmaxf(pos, 0.0f), 15.0f);
  const int lo = (int)floorf(pos);
  const int hi = min(lo + 1, 15);
  const float w = pos - (float)lo;

  const int zbase = 2 + mode * 16;
  const float invre = inv[c * 2], invim = inv[c * 2 + 1];

  float accre = 0.0f, accim = 0.0f;
#pragma unroll 4
  for (int r = 0; r < 16; ++r) {
    const int tb = (((b * 34 + mode) * 16 + c) * 16 + r) * 2;  // shift = trans[:,mode]
    const float shre = trans[tb], shim = trans[tb + 1];
    const int sp = ((c * 16 + r) * 768 + h) * 2;
    const float spre = space[sp] + shre;
    const float spim = space[sp + 1] + shim;
    const int tlo = (((b * 34 + zbase + lo) * 16 + c) * 16 + r) * 2;
    const int thi = (((b * 34 + zbase + hi) * 16 + c) * 16 + r) * 2;
    const float ztre = trans[tlo] * (1.0f - w) + trans[thi] * w;
    const float ztim = trans[tlo + 1] * (1.0f - w) + trans[thi + 1] * w;
    const float sre = spre * ztre - spim * ztim;   // cmul(sp, zt)
    const float sim = spre * ztim + spim * ztre;
    float gre, gim;
    if (mode == 0) {                               // cmul(si, inv)
      gre = sre * invre - sim * invim;
      gim = sre * invim + sim * invre;
    } else {                                       // cdiv(inv, sj, 1e-4)
      const float den = fmaxf(sre * sre + sim * sim, 1e-4f);
      gre = (invre * sre + invim * sim) / den;
      gim = (invim * sre - invre * sim) / den;
    }
    const float bta = beta[(b * 16 + c) * 16 + r];
    accre += bta * gre;
    accim += bta * gim;
  }
  const int o = ((b * 16 + c) * 768 + h) * 2;
  gout[o] = accre;
  gout[o + 1] = accim;
}

// ---------------------------------------------------------------------------
// k6: out[b][h][w] = sum_c (wRe*pc + wIm*ps).
// Tile: 128 w x 8 h per block (grid 6 x 96), 128 threads.
//  - proj streamed from HBM exactly once (b inner, h tiled).
//  - gJ slice (4x16x128 complex = 64 KB) staged in LDS once per block:
//    cuts gJ L2 traffic 8x vs per-row reloads.
//  - gI addresses are wave-uniform -> scalar loads via constant cache.
// Inner math: acc += gIr*u + gIi*v with u = gj.x*pc + gj.y*ps,
//             v = gj.x*ps - gj.y*pc  (6 FMA per element, algebraic minimum).
// ---------------------------------------------------------------------------
__global__ void __launch_bounds__(128)
k6_final(const float* __restrict__ gI,   // [4][16][768][2]
         const float* __restrict__ gJ,   // [4][16][768][2]
         const float* __restrict__ proj, // [16][768][768][2]
         float* __restrict__ out)        // [4][768][768]
{
  __shared__ float2 sGJ[4 * 16 * 128];   // 64 KB

  const int wl = threadIdx.x;            // 0..127
  const int w0 = blockIdx.x * 128;
  const int w  = w0 + wl;
  const int h0 = blockIdx.y * 8;

  const float2* __restrict__ gJ2   = (const float2*)gJ;
  const float2* __restrict__ gI2   = (const float2*)gI;
  const float2* __restrict__ proj2 = (const float2*)proj;

  // stage gJ tile: 8192 float2, 64 per thread
#pragma unroll
  for (int it = 0; it < 64; ++it) {
    const int idx = it * 128 + wl;           // [b][c][wl2] flat
    const int bc   = idx >> 7;               // b*16+c
    const int wl2  = idx & 127;
    sGJ[idx] = gJ2[bc * 768 + w0 + wl2];
  }
  __syncthreads();

  float acc[4][8];
#pragma unroll
  for (int b = 0; b < 4; ++b)
#pragma unroll
    for (int i = 0; i < 8; ++i) acc[b][i] = 0.0f;

  for (int c = 0; c < 16; ++c) {
    float pc[8], ps[8];
#pragma unroll
    for (int i = 0; i < 8; ++i) {
      const float2 p = proj2[(size_t)(c * 768 + h0 + i) * 768 + w];
      const float rden = rsqrtf(p.x * p.x + p.y * p.y + 1e-12f);
      pc[i] = p.x * rden;
      ps[i] = p.y * rden;
    }
#pragma unroll
    for (int b = 0; b < 4; ++b) {
      const float2 gj = sGJ[(b * 16 + c) * 128 + wl];
#pragma unroll
      for (int i = 0; i < 8; ++i) {
        const float2 gi = gI2[(b * 16 + c) * 768 + h0 + i];  // uniform -> s_load
        const float u = gj.x * pc[i] + gj.y * ps[i];
        const float v = gj.x * ps[i] - gj.y * pc[i];
        acc[b][i] += gi.x * u + gi.y * v;
      }
    }
  }

#pragma unroll
  for (int b = 0; b < 4; ++b)
#pragma unroll
    for (int i = 0; i < 8; ++i)
      out[((size_t)b * 768 + h0 + i) * 768 + w] = acc[b][i];
}

// ---------------------------------------------------------------------------
extern "C" void kernel_launch(void* const* d_in, const int* in_sizes, int n_in,
                              void* d_out, int out_size, void* d_ws, size_t ws_size,
                              hipStream_t stream) {
  const float* x          = (const float*)d_in[0];  // [4][2048]
  const float* Wc         = (const float*)d_in[1];  // [1024][2048]
  const float* bc         = (const float*)d_in[2];  // [1024]
  const float* space_i    = (const float*)d_in[3];  // [1][16][16][768][2]
  const float* space_j    = (const float*)d_in[4];  // [1][16][16][768][2]
  const float* inversions = (const float*)d_in[5];  // [1][16][1][1][2]
  const float* G_Re       = (const float*)d_in[6];  // [34][16][256][16]
  const float* G_Im       = (const float*)d_in[7];  // [34][16][256][16]
  const float* w_mix      = (const float*)d_in[8];  // [256][256][2]
  const float* proj       = (const float*)d_in[9];  // [1][16][768][768][2]
  float* out = (float*)d_out;                       // [4][768][768]

  float* ws    = (float*)d_ws;
  float* xt    = ws;             // 4096
  float* xr    = ws + 4096;      // 1024
  float* xi    = ws + 5120;      // 1024
  float* trans = ws + 6144;      // 69632
  float* beta  = ws + 75776;     // 1024
  float* gI    = ws + 76800;     // 98304
  float* gJ    = ws + 175104;    // 98304  (total ~1.1 MB)

  k1_gemm_wmma<<<8, 256, 0, stream>>>(x, Wc, bc, xt);
  k2_siglog<<<4, 256, 0, stream>>>(xt, xr, xi);
  k3_einsum_wmma<<<68, 256, 0, stream>>>(xr, xi, G_Re, G_Im, trans);
  k4_beta<<<16, 64, 0, stream>>>(xr, xi, w_mix, beta);
  k5_spaces<<<dim3(3, 64), 256, 0, stream>>>(space_i, trans, beta, inversions, gI, 0);
  k5_spaces<<<dim3(3, 64), 256, 0, stream>>>(space_j, trans, beta, inversions, gJ, 1);
  k6_final<<<dim3(6, 96), 128, 0, stream>>>(gI, gJ, proj, out);
}